// RobustGlobalPool2d_35029753266606
// MI455X (gfx1250) — compile-verified
//
#include <hip/hip_runtime.h>
#include <hip/hip_bf16.h>

#ifndef __has_builtin
#define __has_builtin(x) 0
#endif

#if __has_builtin(__builtin_amdgcn_global_load_async_to_lds_b128)
#define HAS_ASYNC_LDS 1
#else
#define HAS_ASYNC_LDS 0
#endif

#define NEWTON_ITERS 30
#define ROW_N        16384    // H*W = 128*128
#define TPB          256      // 8 wave32 waves
#define F4T          16       // float4 per thread -> 64 floats/thread

#define AS1 __attribute__((address_space(1)))
#define AS3 __attribute__((address_space(3)))

typedef int v4i __attribute__((vector_size(16)));   // matches builtin param type

__device__ __forceinline__ void wait_asynccnt_zero() {
#if __has_builtin(__builtin_amdgcn_s_wait_asynccnt)
    __builtin_amdgcn_s_wait_asynccnt(0);
#else
    asm volatile("s_wait_asynccnt 0" ::: "memory");
#endif
}

__global__ __launch_bounds__(TPB) void RobustGlobalPool2d_kernel(
    const float* __restrict__ x,
    const float* __restrict__ scale,
    float* __restrict__ out)
{
#if HAS_ASYNC_LDS
    __shared__ float4 tile[ROW_N / 4];      // 64 KB: staged row (async DMA target)
    float* redf = (float*)tile;             // reused as reduction scratch afterwards
#else
    __shared__ float redbuf[2 * (TPB / 32)];
    float* redf = redbuf;
#endif
    const int    t    = threadIdx.x;
    const int    wid  = t >> 5;
    const int    lane = t & 31;
    const size_t row  = blockIdx.x;
    const float4* __restrict__ xrow = (const float4*)(x + row * (size_t)ROW_N);

#if HAS_ASYNC_LDS
    // Stage this row global -> LDS with gfx1250 async copies (ASYNCcnt-tracked).
    // Each thread DMAs exactly the float4s it will later consume, so only its
    // own asynccnt wait is required (no workgroup barrier before readback).
    #pragma unroll
    for (int k = 0; k < F4T; ++k) {
        const int f = k * TPB + t;          // coalesced: wave covers 512B/instr
        __builtin_amdgcn_global_load_async_to_lds_b128(
            (AS1 v4i*)(xrow + f), (AS3 v4i*)(tile + f),
            /*imm offset*/0, /*cpol*/0);
    }
    wait_asynccnt_zero();
#endif

    // Pull own 64 elements into registers; accumulate sum for the mean init.
    float4 v[F4T];
    float lsum = 0.0f;
    #pragma unroll
    for (int k = 0; k < F4T; ++k) {
        const int f = k * TPB + t;
#if HAS_ASYNC_LDS
        v[k] = tile[f];                     // ds_load_b128, conflict-free
#else
        v[k] = xrow[f];
#endif
        lsum += (v[k].x + v[k].y) + (v[k].z + v[k].w);
    }

    // ---- y0 = mean(row) ----
    #pragma unroll
    for (int off = 16; off; off >>= 1) lsum += __shfl_xor(lsum, off, 32);
    __syncthreads();                         // tile consumed; safe to reuse as redf
    if (lane == 0) redf[wid] = lsum;
    __syncthreads();
    float tot = 0.0f;
    #pragma unroll
    for (int w = 0; w < TPB / 32; ++w) tot += redf[w];
    float y = tot * (1.0f / (float)ROW_N);

    const float inv_s = 1.0f / scale[0];

    // ---- 30 Newton steps on the pseudo-Huber objective ----
    #pragma unroll 1
    for (int it = 0; it < NEWTON_ITERS; ++it) {
        float g = 0.0f, h = 0.0f;
        #pragma unroll
        for (int k = 0; k < F4T; ++k) {
            const float e[4] = { v[k].x, v[k].y, v[k].z, v[k].w };
            #pragma unroll
            for (int c = 0; c < 4; ++c) {
                const float z  = y - e[c];
                const float zs = z * inv_s;
                const float tt = fmaf(zs, zs, 1.0f);     // 1 + (z/s)^2 >= 1
                const float rs = __builtin_amdgcn_rsqf(tt); // v_rsq_f32
                g = fmaf(z, rs, g);                      // z * t^-1/2
                h = fmaf(rs * rs, rs, h);                // t^-3/2
            }
        }
        #pragma unroll
        for (int off = 16; off; off >>= 1) {
            g += __shfl_xor(g, off, 32);
            h += __shfl_xor(h, off, 32);
        }
        __syncthreads();                     // prior iteration's redf reads done
        if (lane == 0) { redf[2 * wid] = g; redf[2 * wid + 1] = h; }
        __syncthreads();
        float G = 0.0f, H = 0.0f;
        #pragma unroll
        for (int w = 0; w < TPB / 32; ++w) { G += redf[2 * w]; H += redf[2 * w + 1]; }
        y -= G / H;                          // uniform across block
    }

    if (t == 0) out[row] = y;
}

extern "C" void kernel_launch(void* const* d_in, const int* in_sizes, int n_in,
                              void* d_out, int out_size, void* d_ws, size_t ws_size,
                              hipStream_t stream) {
    (void)n_in; (void)out_size; (void)d_ws; (void)ws_size;
    const float* x     = (const float*)d_in[0];
    const float* scale = (const float*)d_in[1];
    float*       out   = (float*)d_out;
    const int rows = in_sizes[0] / ROW_N;    // B*C = 4096
    RobustGlobalPool2d_kernel<<<dim3(rows), dim3(TPB), 0, stream>>>(x, scale, out);
}